// StaticFusionEncoder_40243843564282
// MI455X (gfx1250) — compile-verified
//
#include <hip/hip_runtime.h>
#include <math.h>

typedef float v2f __attribute__((ext_vector_type(2)));
typedef float v8f __attribute__((ext_vector_type(8)));
typedef unsigned int v4u __attribute__((ext_vector_type(4)));
typedef unsigned int v8u __attribute__((ext_vector_type(8)));

namespace {
constexpr int  kD   = 10;
constexpr int  kH   = 192;
constexpr long kRows = 8L * 65536L;            // B*P = 524288
constexpr int  kWaves = 8;
constexpr int  kRowsPerBlock = kWaves * 16;    // 128 rows per workgroup

// ---- LDS layout (bytes). All v2f regions 8-byte aligned. ----
constexpr int OFF_FC1W = 0;                    // v2f  [6][192]   (K padded 10->12, packed (k,k+1))
constexpr int OFF_FC1B = OFF_FC1W + 6 * 192 * 8;   // float[192]
constexpr int OFF_FC2B = OFF_FC1B + 192 * 4;       // float[192]
constexpr int OFF_FC2W = OFF_FC2B + 192 * 4;       // float[192][192] row-major (TDM target)
constexpr int OFF_H    = OFF_FC2W + 192 * 192 * 4; // float[8][16][192]  per-wave h tiles
constexpr int OFF_MASK = OFF_H + 8 * 16 * 192 * 4; // float[8][16]
constexpr int kLdsBytes = OFF_MASK + 8 * 16 * 4;   // 257024 B  (< 320 KB WGP limit)

constexpr unsigned kFc2Elems = 192u * 192u;        // 36864 (< 65536, fits tile_dim0)
} // namespace

__global__ __launch_bounds__(256, 1) void encoder_fused(
    const float* __restrict__ x,
    const float* __restrict__ fc1w, const float* __restrict__ fc1b,
    const float* __restrict__ fc2w, const float* __restrict__ fc2b,
    float* __restrict__ out, float* __restrict__ maskout, float* __restrict__ posout)
{
    extern __shared__ char smem[];
    v2f*   s_fc1w = reinterpret_cast<v2f*>(smem + OFF_FC1W);
    float* s_fc1b = reinterpret_cast<float*>(smem + OFF_FC1B);
    float* s_fc2b = reinterpret_cast<float*>(smem + OFF_FC2B);
    float* s_fc2w = reinterpret_cast<float*>(smem + OFF_FC2W);
    float* s_h    = reinterpret_cast<float*>(smem + OFF_H);
    float* s_mask = reinterpret_cast<float*>(smem + OFF_MASK);

    const int  tid  = threadIdx.x;
    const int  wave = tid >> 5;
    const int  lane = tid & 31;
    const int  l16  = lane & 15;
    const bool hi   = lane >= 16;   // upper half-wave handles K+2/K+3 (A) and M+8 (C/D)

    // ---------------- TDM: DMA fc2_w (147456 B) straight into LDS from wave 0 ----------------
    // D# describes a 1-D tensor of 36864 f32 elements; tile == tensor. Overlaps with the
    // cooperative fc1 staging below. Tracked by TENSORcnt.
    if (wave == 0) {
        const unsigned long long ga = (unsigned long long)(const void*)fc2w;
        v4u g0;
        g0.x = 1u;                                     // count=1, is_restore=0, gather off
        g0.y = (unsigned)OFF_FC2W;                     // lds_addr (bytes)
        g0.z = (unsigned)(ga & 0xffffffffu);           // global_addr[31:0]
        g0.w = (unsigned)((ga >> 32) & 0x01ffffffu)    // global_addr[56:32]
             | 0x80000000u;                            // type = 2 ("image")
        v8u g1;
        g1[0] = (2u << 16);                            // workgroup_mask=0, data_size=4B
        g1[1] = (kFc2Elems & 0xffffu) << 16;           // tensor_dim0 low16 (bits 63:48)
        g1[2] = ((kFc2Elems >> 16) & 0xffffu)          // tensor_dim0 high16 (bits 79:64)
              | (1u << 16);                            // tensor_dim1 = 1 (bits 95:80)
        g1[3] = (kFc2Elems & 0xffffu) << 16;           // tile_dim0 (bits 127:112)
        g1[4] = 0u;                                    // tile_dim1/2 unused
        g1[5] = kFc2Elems;                             // tensor_dim0_stride low32
        g1[6] = 0u;
        g1[7] = 0u;
        v4u g2; g2.x = 1u; g2.y = 1u; g2.z = 0u; g2.w = 0u;   // tensor_dim2/3 = 1, tile_dim3 unused
        v4u g3; g3.x = 0u; g3.y = 0x00010000u; g3.z = 0u; g3.w = 0u; // tensor_dim4 = 1, tile_dim4 unused
        asm volatile("tensor_load_to_lds %0, %1, %2, %3"
                     :: "s"(g0), "s"(g1), "s"(g2), "s"(g3)
                     : "memory");
        __builtin_amdgcn_s_wait_tensorcnt(0);          // wave 0 waits before signaling barrier
    }

    // ---------------- Stage fc1 weights (packed (k,k+1) pairs) + biases ----------------
    for (int i = tid; i < 6 * 192; i += 256) {
        const int kp = i / 192, n = i - kp * 192;
        v2f w;
        w.x = (2 * kp     < kD) ? fc1w[(2 * kp    ) * kH + n] : 0.0f;
        w.y = (2 * kp + 1 < kD) ? fc1w[(2 * kp + 1) * kH + n] : 0.0f;
        s_fc1w[i] = w;
    }
    for (int i = tid; i < kH; i += 256) { s_fc1b[i] = fc1b[i]; s_fc2b[i] = fc2b[i]; }
    __syncthreads();

    const long rowbase = (long)blockIdx.x * kRowsPerBlock + (long)wave * 16;
    const long myrow   = rowbase + l16;               // this lane's A-matrix row (M = lane%16)
    const float* xrow  = x + myrow * kD;

    // Prefetch next block's slice of x into the cache hierarchy (global_prefetch_b8).
    __builtin_prefetch(x + (rowbase + kRowsPerBlock) * kD, 0, 1);

    // ---------------- GEMM1 A-fragments (16x4 f32 tiles, K padded to 12) ----------------
    // Lane<16 holds (K+0,K+1) of row l16; lane>=16 holds (K+2,K+3). 8-byte aligned loads.
    v2f a1[3];
    a1[0] = *reinterpret_cast<const v2f*>(xrow + (hi ? 2 : 0));
    a1[1] = *reinterpret_cast<const v2f*>(xrow + (hi ? 6 : 4));
    v2f zz; zz.x = 0.0f; zz.y = 0.0f;
    a1[2] = hi ? zz : *reinterpret_cast<const v2f*>(xrow + 8);   // K=10,11 are zero pad

    // ---------------- mask + pos side outputs (one row per lane, lanes 0-15) ----------------
    if (!hi) {
        const v2f x23 = *reinterpret_cast<const v2f*>(xrow + 2);
        const v2f x67 = *reinterpret_cast<const v2f*>(xrow + 6);
        const int nz = (a1[0].x != 0.0f) + (a1[0].y != 0.0f) +
                       (x23.x   != 0.0f) + (x23.y   != 0.0f) +
                       (a1[1].x != 0.0f) + (a1[1].y != 0.0f) +
                       (x67.x   != 0.0f) + (x67.y   != 0.0f) +
                       (a1[2].x != 0.0f) + (a1[2].y != 0.0f);
        const float m = (nz == 0) ? 1.0f : 0.0f;
        s_mask[wave * 16 + l16] = m;
        maskout[myrow] = m;
        float* pr = posout + myrow * 7;
        pr[0] = a1[0].x; pr[1] = a1[0].y; pr[2] = x23.x; pr[3] = x23.y;
        pr[4] = 0.0f;    pr[5] = 1.0f;    pr[6] = 0.0f;
    }

    // ---------------- GEMM1: h = gelu(x @ fc1_w + b1), staged to per-wave LDS tile ----------------
    float* htile = s_h + wave * 16 * kH;             // [16][192]
    for (int n = 0; n < 12; ++n) {
        v8f c = {};
        #pragma unroll
        for (int kb = 0; kb < 3; ++kb) {
            const v2f b = s_fc1w[(2 * kb + (hi ? 1 : 0)) * kH + n * 16 + l16];
            c = __builtin_amdgcn_wmma_f32_16x16x4_f32(
                    false, a1[kb], false, b, (short)0, c, false, false);
        }
        const float bias = s_fc1b[n * 16 + l16];
        #pragma unroll
        for (int r = 0; r < 8; ++r) {
            float v = c[r] + bias;
            v = 0.5f * v * (1.0f + erff(v * 0.70710678118654752f));   // exact gelu
            const int mrow = r + (hi ? 8 : 0);
            htile[mrow * kH + n * 16 + l16] = v;
        }
    }
    // Same-wave LDS producer->consumer: DScnt waits cover this; no workgroup barrier needed.

    // ---------------- GEMM2 A-fragments: 48 K-blocks preloaded into registers ----------------
    const float* hrow = htile + l16 * kH;            // A row M = lane%16
    v2f a2[48];
    #pragma unroll
    for (int kb = 0; kb < 48; ++kb)
        a2[kb] = *reinterpret_cast<const v2f*>(hrow + 4 * kb + (hi ? 2 : 0));

    // ---------------- GEMM2: out = (h @ fc2_w + b2) * (1 - mask) ----------------
    for (int n = 0; n < 12; ++n) {
        v8f c = {};
        const int col = n * 16 + l16;
        #pragma unroll
        for (int kb = 0; kb < 48; ++kb) {
            const int k0 = 4 * kb + (hi ? 2 : 0);
            v2f b;                                    // (k0, k0+1) rows, same col:
            b.x = s_fc2w[(k0    ) * kH + col];        // 192-dword delta -> ds_load_2addr_b32
            b.y = s_fc2w[(k0 + 1) * kH + col];
            c = __builtin_amdgcn_wmma_f32_16x16x4_f32(
                    false, a2[kb], false, b, (short)0, c, false, false);
        }
        const float bias = s_fc2b[col];
        #pragma unroll
        for (int r = 0; r < 8; ++r) {
            const int  mrow = r + (hi ? 8 : 0);
            const float m  = s_mask[wave * 16 + mrow];        // 1.0 -> zero the row
            const float v  = (c[r] + bias) * (1.0f - m);
            out[(rowbase + mrow) * kH + col] = v;
        }
    }
}

extern "C" void kernel_launch(void* const* d_in, const int* in_sizes, int n_in,
                              void* d_out, int out_size, void* d_ws, size_t ws_size,
                              hipStream_t stream) {
    (void)in_sizes; (void)n_in; (void)out_size; (void)d_ws; (void)ws_size;
    const float* x    = (const float*)d_in[0];
    const float* fc1w = (const float*)d_in[1];
    const float* fc1b = (const float*)d_in[2];
    const float* fc2w = (const float*)d_in[3];
    const float* fc2b = (const float*)d_in[4];

    float* out     = (float*)d_out;                 // (B,P,192)
    float* maskout = out + kRows * kH;              // (B,P) as 0/1 float
    float* posout  = maskout + kRows;               // (B,P,7)

    hipFuncSetAttribute(reinterpret_cast<const void*>(encoder_fused),
                        hipFuncAttributeMaxDynamicSharedMemorySize, kLdsBytes);

    dim3 grid((unsigned)(kRows / kRowsPerBlock));   // 4096 blocks
    encoder_fused<<<grid, 256, kLdsBytes, stream>>>(x, fc1w, fc1b, fc2w, fc2b,
                                                    out, maskout, posout);
}